// SAM_14061722927495
// MI455X (gfx1250) — compile-verified
//
#include <hip/hip_runtime.h>

// ---- problem constants --------------------------------------------------
#define B_     8
#define CIN_   256
#define REL_   16
#define OUT_   64
#define SH_    8
#define H_     56
#define W_     56
#define HW_    (H_ * W_)            // 3136
#define KWIN   7
#define NCAT   (REL_ * (KWIN*KWIN + 1))   // 800
#define NWM    (KWIN*KWIN * (OUT_/SH_))   // 392
#define NT_PER_B (HW_ / 16)         // 196
#define NTILES (B_ * NT_PER_B)      // 1568

typedef __attribute__((ext_vector_type(2))) float v2f;
typedef __attribute__((ext_vector_type(8))) float v8f;

// fp32 WMMA: D(16x16,f32) = A(16x4,f32) x B(4x16,f32) + C
// A lane layout: M = lane&15, K = 2*(lane>>4) + v   (v2f)
// B lane layout: N = lane&15, K = 2*(lane>>4) + v   (v2f)
// C/D layout:    N = lane&15, M = v + 8*(lane>>4)   (v8f)
__device__ __forceinline__ v8f wmma_f32(v2f a, v2f b, v8f c) {
    return __builtin_amdgcn_wmma_f32_16x16x4_f32(false, a, false, b,
                                                 (short)0, c, false, false);
}

// ---- workspace layout (floats) -----------------------------------------
#define WS_X1   0                                   // B*16*HW
#define WS_X2   (WS_X1 + B_*REL_*HW_)               // B*16*HW
#define WS_X3   (WS_X2 + B_*REL_*HW_)               // B*64*HW
#define WS_T    (WS_X3 + B_*OUT_*HW_)               // B*8*HW  (already bn2+relu)
#define WS_BN   (WS_T  + B_*(OUT_/SH_)*HW_)         // 800 scale,800 shift,8 scale,8 shift
#define WS_LUT  (WS_BN + 2*NCAT + 2*(OUT_/SH_))     // 800 packed ints (gather LUT)

// =========================================================================
// K0: fold BN params into scale/shift + build the cat-channel gather LUT
// LUT code: bit30 set -> x1 channel (low 8 bits)
//           else       r = bits[7:0], 2i = bits[15:8], 2j = bits[23:16]
// =========================================================================
__global__ void sam_bnprep(const float* __restrict__ g1, const float* __restrict__ b1,
                           const float* __restrict__ m1, const float* __restrict__ v1,
                           const float* __restrict__ g2, const float* __restrict__ b2,
                           const float* __restrict__ m2, const float* __restrict__ v2,
                           float* __restrict__ bnS, int* __restrict__ lut) {
    int i = blockIdx.x * blockDim.x + threadIdx.x;
    if (i < NCAT) {
        float s = g1[i] * rsqrtf(v1[i] + 1e-5f);
        bnS[i]          = s;
        bnS[NCAT + i]   = b1[i] - m1[i] * s;
        int code;
        if (i < REL_) {
            code = (1 << 30) | i;
        } else {
            int cc = i - REL_;
            int r  = cc / 49;
            int k  = cc - r * 49;
            int ii = k / KWIN;
            int jj = k - ii * KWIN;
            code = r | ((2 * ii) << 8) | ((2 * jj) << 16);
        }
        lut[i] = code;
    }
    if (i < OUT_/SH_) {
        float s = g2[i] * rsqrtf(v2[i] + 1e-5f);
        bnS[2*NCAT + i]              = s;
        bnS[2*NCAT + OUT_/SH_ + i]   = b2[i] - m2[i] * s;
    }
}

// =========================================================================
// K1: fused QKV 1x1 convs.  y[96, B*HW] = W[96,256] @ X[256, B*HW] + bias
// one wave per (m-tile of 6, 64-column group): A fragment reused across
// 4 N-tiles -> 4 independent WMMAs per K-step, 4x less weight traffic.
// =========================================================================
__global__ __launch_bounds__(256) void sam_qkv(
    const float* __restrict__ x,
    const float* __restrict__ w1, const float* __restrict__ b1,
    const float* __restrict__ w2, const float* __restrict__ b2,
    const float* __restrict__ w3, const float* __restrict__ b3,
    float* __restrict__ ws) {
    int wid  = (blockIdx.x * blockDim.x + threadIdx.x) >> 5;   // 2352 waves
    int lane = threadIdx.x & 31;
    int lm = lane & 15, half = lane >> 4;

    int mt  = wid % 6;
    int ng  = wid / 6;                    // 64-column group: 0..391
    int b   = ng / (NT_PER_B / 4);        // 49 groups per batch
    int n0  = (ng % (NT_PER_B / 4)) * 64;

    const float* wp; const float* bp; float* op; int mrow;
    if (mt == 0)      { wp = w1; bp = b1; op = ws + WS_X1 + (size_t)b*REL_*HW_; mrow = 0; }
    else if (mt == 1) { wp = w2; bp = b2; op = ws + WS_X2 + (size_t)b*REL_*HW_; mrow = 0; }
    else              { wp = w3; bp = b3; op = ws + WS_X3 + (size_t)b*OUT_*HW_; mrow = (mt-2)*16; }

    const float* xb   = x + (size_t)b * CIN_ * HW_ + n0 + lm;   // column base
    const float* wrow = wp + (size_t)(mrow + lm) * CIN_;        // A row base

    v8f acc0 = {}, acc1 = {}, acc2 = {}, acc3 = {};
    #pragma unroll 2
    for (int k0 = 0; k0 < CIN_; k0 += 4) {
        int k = k0 + half * 2;
        v2f a;
        a.x = wrow[k];
        a.y = wrow[k + 1];
        const float* r0 = xb + (size_t)k * HW_;
        const float* r1 = xb + (size_t)(k + 1) * HW_;
        v2f bf0, bf1, bf2, bf3;
        bf0.x = r0[0];  bf0.y = r1[0];
        bf1.x = r0[16]; bf1.y = r1[16];
        bf2.x = r0[32]; bf2.y = r1[32];
        bf3.x = r0[48]; bf3.y = r1[48];
        acc0 = wmma_f32(a, bf0, acc0);
        acc1 = wmma_f32(a, bf1, acc1);
        acc2 = wmma_f32(a, bf2, acc2);
        acc3 = wmma_f32(a, bf3, acc3);
    }
    #pragma unroll
    for (int v = 0; v < 8; ++v) {
        int M = mrow + v + half * 8;                 // row in this conv's output
        float bias = bp[M];
        float* orow = op + (size_t)M * HW_ + n0 + lm;
        orow[0]  = acc0[v] + bias;
        orow[16] = acc1[v] + bias;
        orow[32] = acc2[v] + bias;
        orow[48] = acc3[v] + bias;
    }
}

// =========================================================================
// K2: t = relu(bn2( wcw1 @ relu(bn1(cat)) ))
// cat channel c: LUT-decoded gather; reflect pad 2 + zero pad 4, 7x7, dil 2.
// Zero-padded taps still pass through BN+ReLU (relu(shift) != 0 in general).
// =========================================================================
__device__ __forceinline__ float sam_catval(const float* __restrict__ x1,
                                            const float* __restrict__ x2,
                                            const float* __restrict__ bnS,
                                            const int* __restrict__ lut,
                                            int b, int c, int y, int x) {
    int code = lut[c];
    float raw = 0.0f;
    if (code & (1 << 30)) {
        raw = x1[((size_t)b * REL_ + (code & 0xff)) * HW_ + y * W_ + x];
    } else {
        int r  = code & 0xff;
        int sy = y + ((code >> 8)  & 0xff) - 6;
        int sx = x + ((code >> 16) & 0xff) - 6;
        bool inside = (sy >= -2) && (sy <= H_ + 1) && (sx >= -2) && (sx <= W_ + 1);
        sy = (sy < 0) ? -sy : ((sy >= H_) ? 2 * (H_ - 1) - sy : sy);
        sx = (sx < 0) ? -sx : ((sx >= W_) ? 2 * (W_ - 1) - sx : sx);
        if (inside)
            raw = x2[((size_t)b * REL_ + r) * HW_ + sy * W_ + sx];
    }
    return fmaxf(fmaf(raw, bnS[c], bnS[NCAT + c]), 0.0f);
}

__global__ __launch_bounds__(256) void sam_attn1(
    const float* __restrict__ wcw1, float* __restrict__ ws) {
    const float* x1  = ws + WS_X1;
    const float* x2  = ws + WS_X2;
    const float* bnS = ws + WS_BN;
    const int*   lut = (const int*)(ws + WS_LUT);
    float*       t   = ws + WS_T;

    int wid  = (blockIdx.x * blockDim.x + threadIdx.x) >> 5;   // 1568 waves
    int lane = threadIdx.x & 31;
    int lm = lane & 15, half = lane >> 4;

    int b  = wid / NT_PER_B;
    int n0 = (wid % NT_PER_B) * 16;
    int n  = n0 + lm;
    int y  = n / W_, xx = n - y * W_;

    const float* arow = wcw1 + (size_t)lm * NCAT;   // valid rows 0..7
    bool arv = (lm < (OUT_/SH_));

    v8f acc = {};
    for (int c0 = 0; c0 < NCAT; c0 += 4) {
        int c = c0 + half * 2;
        v2f a, bf;
        a.x  = arv ? arow[c]     : 0.0f;
        a.y  = arv ? arow[c + 1] : 0.0f;
        bf.x = sam_catval(x1, x2, bnS, lut, b, c,     y, xx);
        bf.y = sam_catval(x1, x2, bnS, lut, b, c + 1, y, xx);
        acc = wmma_f32(a, bf, acc);
    }
    if (half == 0) {                 // rows 0..7 live in half 0
        #pragma unroll
        for (int v = 0; v < 8; ++v) {
            float s  = bnS[2*NCAT + v];
            float sh = bnS[2*NCAT + OUT_/SH_ + v];
            t[((size_t)b * (OUT_/SH_) + v) * HW_ + n] = fmaxf(fmaf(acc[v], s, sh), 0.0f);
        }
    }
}

// =========================================================================
// K3: fused  wmap = wcw2[392,8] @ t  (+bias)  ->  LDS  ->  49-tap aggregation
// one block (8 waves) per 16-column tile
// =========================================================================
__global__ __launch_bounds__(256) void sam_attn2_agg(
    const float* __restrict__ wcw2, const float* __restrict__ bcw2,
    const float* __restrict__ ws_ro, float* __restrict__ out) {
    __shared__ float wl[NWM * 16];                  // 392 x 16 wmap tile (25 KB)

    const float* t  = ws_ro + WS_T;
    const float* x3 = ws_ro + WS_X3;

    int b  = blockIdx.x / NT_PER_B;
    int n0 = (blockIdx.x % NT_PER_B) * 16;

    int w    = threadIdx.x >> 5;
    int lane = threadIdx.x & 31;
    int lm = lane & 15, half = lane >> 4;

    const float* tb = t + (size_t)b * (OUT_/SH_) * HW_ + n0 + lm;
    for (int mt = w; mt < 25; mt += 8) {            // 25 m-tiles over 8 waves
        int m   = mt * 16 + lm;
        bool mv = (m < NWM);
        const float* arow = wcw2 + (size_t)(mv ? m : 0) * SH_;
        v8f acc = {};
        #pragma unroll
        for (int k0 = 0; k0 < SH_; k0 += 4) {
            int k = k0 + half * 2;
            v2f a, bf;
            a.x  = mv ? arow[k]     : 0.0f;
            a.y  = mv ? arow[k + 1] : 0.0f;
            bf.x = tb[(size_t)k * HW_];
            bf.y = tb[(size_t)(k + 1) * HW_];
            acc = wmma_f32(a, bf, acc);
        }
        #pragma unroll
        for (int v = 0; v < 8; ++v) {
            int M = mt * 16 + v + half * 8;
            if (M < NWM) wl[M * 16 + lm] = acc[v] + bcw2[M];
        }
    }
    __syncthreads();

    // aggregation: 64 channels x 16 columns, 4 consecutive channels per thread
    int col = threadIdx.x & 15;
    int ch0 = (threadIdx.x >> 4) * 4;               // 0..60, same group g for all 4
    int g   = ch0 >> 3;
    int n   = n0 + col;
    int y   = n / W_, xx = n - y * W_;

    const float* x3b = x3 + ((size_t)b * OUT_ + ch0) * HW_;
    float s0 = 0.f, s1 = 0.f, s2 = 0.f, s3 = 0.f;
    #pragma unroll
    for (int i = 0; i < KWIN; ++i) {
        int sy = y + 2 * i - 6;
        sy = (sy < 0) ? -sy : ((sy >= H_) ? 2 * (H_ - 1) - sy : sy);
        #pragma unroll
        for (int j = 0; j < KWIN; ++j) {
            int sx = xx + 2 * j - 6;
            sx = (sx < 0) ? -sx : ((sx >= W_) ? 2 * (W_ - 1) - sx : sx);
            float wv = wl[(g * 49 + i * KWIN + j) * 16 + col];
            size_t off = (size_t)sy * W_ + sx;
            s0 = fmaf(x3b[off],            wv, s0);
            s1 = fmaf(x3b[HW_ + off],      wv, s1);
            s2 = fmaf(x3b[2 * HW_ + off],  wv, s2);
            s3 = fmaf(x3b[3 * HW_ + off],  wv, s3);
        }
    }
    float* ob = out + ((size_t)b * OUT_ + ch0) * HW_ + n;
    ob[0]       = s0;
    ob[HW_]     = s1;
    ob[2 * HW_] = s2;
    ob[3 * HW_] = s3;
}

// =========================================================================
extern "C" void kernel_launch(void* const* d_in, const int* in_sizes, int n_in,
                              void* d_out, int out_size, void* d_ws, size_t ws_size,
                              hipStream_t stream) {
    const float* x     = (const float*)d_in[0];
    const float* w1    = (const float*)d_in[1];
    const float* b1    = (const float*)d_in[2];
    const float* w2    = (const float*)d_in[3];
    const float* b2    = (const float*)d_in[4];
    const float* w3    = (const float*)d_in[5];
    const float* b3    = (const float*)d_in[6];
    const float* bn1g  = (const float*)d_in[7];
    const float* bn1b  = (const float*)d_in[8];
    const float* bn1m  = (const float*)d_in[9];
    const float* bn1v  = (const float*)d_in[10];
    const float* wcw1  = (const float*)d_in[11];
    const float* bn2g  = (const float*)d_in[12];
    const float* bn2b  = (const float*)d_in[13];
    const float* bn2m  = (const float*)d_in[14];
    const float* bn2v  = (const float*)d_in[15];
    const float* wcw2  = (const float*)d_in[16];
    const float* bcw2  = (const float*)d_in[17];
    float* out = (float*)d_out;
    float* ws  = (float*)d_ws;

    sam_bnprep<<<4, 256, 0, stream>>>(bn1g, bn1b, bn1m, bn1v,
                                      bn2g, bn2b, bn2m, bn2v,
                                      ws + WS_BN, (int*)(ws + WS_LUT));

    // K1: 6 m-tiles * 392 col-groups = 2352 waves = 294 blocks of 8 waves
    sam_qkv<<<(6 * NTILES / 4) / 8, 256, 0, stream>>>(x, w1, b1, w2, b2, w3, b3, ws);

    // K2: 1568 waves = 196 blocks
    sam_attn1<<<NTILES / 8, 256, 0, stream>>>(wcw1, ws);

    // K3: one block per n-tile
    sam_attn2_agg<<<NTILES, 256, 0, stream>>>(wcw2, bcw2, ws, out);
}